// MultiheadAttentionLocal_77945066487883
// MI455X (gfx1250) — compile-verified
//
#include <hip/hip_runtime.h>
#include <math.h>

#define EMBED 256
#define KNBR  32
#define HEADS 8
#define HDIM  32

typedef float v2f __attribute__((ext_vector_type(2)));
typedef float v4f __attribute__((ext_vector_type(4)));
typedef float v8f __attribute__((ext_vector_type(8)));

// Y[n_rows x n_cols] = scale * (X[n_rows x k_dim] @ W[n_cols x k_dim]^T + bias)
//
// One wave -> 16x64 output strip (4 f32 WMMA accumulators sharing the A frag).
// All 8 waves of a block share one 64-col group (B frags identical -> L0 hits)
// and cover 8 consecutive 16-row M tiles.
//
// K is consumed 8 logical k's per b128 load: lane-half 0 carries k..k+3,
// lane-half 1 carries k+4..k+7. WMMA step "lo" uses .xy (HW slots K0/K1 for
// lanes 0-15, K2/K3 for lanes 16-31), step "hi" uses .zw. A and B use the same
// logical->slot mapping, so the contraction is exact. Next iteration's five
// fragments are prefetched before the 8 WMMAs to keep loads in flight.
__global__ __launch_bounds__(256) void gemm_xwt_wmma(
    const float* __restrict__ X, const float* __restrict__ W,
    const float* __restrict__ bias, float* __restrict__ Y,
    int n_rows, int k_dim, int n_cols, float scale)
{
    const int lane = threadIdx.x & 31;
    const int wave = threadIdx.x >> 5;
    const int half = lane >> 4;   // 0: lanes 0-15, 1: lanes 16-31
    const int l16  = lane & 15;

    const int colgroups = n_cols >> 6;                 // 64-wide column groups
    const int cg     = blockIdx.x % colgroups;
    const int tile_m = (blockIdx.x / colgroups) * (blockDim.x >> 5) + wave;
    if (tile_m * 16 >= n_rows) return;                 // wave-uniform

    const int row0 = tile_m * 16;
    const int col0 = cg * 64;

    v8f acc0 = {}, acc1 = {}, acc2 = {}, acc3 = {};

    const float* aP  = X + (size_t)(row0 + l16)      * k_dim + half * 4;
    const float* bP0 = W + (size_t)(col0 +  0 + l16) * k_dim + half * 4;
    const float* bP1 = W + (size_t)(col0 + 16 + l16) * k_dim + half * 4;
    const float* bP2 = W + (size_t)(col0 + 32 + l16) * k_dim + half * 4;
    const float* bP3 = W + (size_t)(col0 + 48 + l16) * k_dim + half * 4;

#define WMMA_STEP8(AV, B0V, B1V, B2V, B3V)                                          \
    {                                                                               \
        v2f alo = __builtin_shufflevector(AV, AV, 0, 1);                            \
        v2f ahi = __builtin_shufflevector(AV, AV, 2, 3);                            \
        v2f blo, bhi;                                                               \
        blo = __builtin_shufflevector(B0V, B0V, 0, 1);                              \
        bhi = __builtin_shufflevector(B0V, B0V, 2, 3);                              \
        acc0 = __builtin_amdgcn_wmma_f32_16x16x4_f32(false, alo, false, blo, (short)0, acc0, false, false); \
        acc0 = __builtin_amdgcn_wmma_f32_16x16x4_f32(false, ahi, false, bhi, (short)0, acc0, false, false); \
        blo = __builtin_shufflevector(B1V, B1V, 0, 1);                              \
        bhi = __builtin_shufflevector(B1V, B1V, 2, 3);                              \
        acc1 = __builtin_amdgcn_wmma_f32_16x16x4_f32(false, alo, false, blo, (short)0, acc1, false, false); \
        acc1 = __builtin_amdgcn_wmma_f32_16x16x4_f32(false, ahi, false, bhi, (short)0, acc1, false, false); \
        blo = __builtin_shufflevector(B2V, B2V, 0, 1);                              \
        bhi = __builtin_shufflevector(B2V, B2V, 2, 3);                              \
        acc2 = __builtin_amdgcn_wmma_f32_16x16x4_f32(false, alo, false, blo, (short)0, acc2, false, false); \
        acc2 = __builtin_amdgcn_wmma_f32_16x16x4_f32(false, ahi, false, bhi, (short)0, acc2, false, false); \
        blo = __builtin_shufflevector(B3V, B3V, 0, 1);                              \
        bhi = __builtin_shufflevector(B3V, B3V, 2, 3);                              \
        acc3 = __builtin_amdgcn_wmma_f32_16x16x4_f32(false, alo, false, blo, (short)0, acc3, false, false); \
        acc3 = __builtin_amdgcn_wmma_f32_16x16x4_f32(false, ahi, false, bhi, (short)0, acc3, false, false); \
    }

    v4f a  = *(const v4f*)(aP);
    v4f b0 = *(const v4f*)(bP0);
    v4f b1 = *(const v4f*)(bP1);
    v4f b2 = *(const v4f*)(bP2);
    v4f b3 = *(const v4f*)(bP3);

    for (int k = 8; k < k_dim; k += 8) {
        // prefetch next iteration's fragments (stay in flight over the WMMAs)
        v4f an  = *(const v4f*)(aP  + k);
        v4f bn0 = *(const v4f*)(bP0 + k);
        v4f bn1 = *(const v4f*)(bP1 + k);
        v4f bn2 = *(const v4f*)(bP2 + k);
        v4f bn3 = *(const v4f*)(bP3 + k);

        WMMA_STEP8(a, b0, b1, b2, b3);

        a = an; b0 = bn0; b1 = bn1; b2 = bn2; b3 = bn3;
    }
    WMMA_STEP8(a, b0, b1, b2, b3);
#undef WMMA_STEP8

    // C/D layout: vgpr r -> M = r + 8*(lane/16), N = lane%16
    v8f accs[4] = {acc0, acc1, acc2, acc3};
    #pragma unroll
    for (int c = 0; c < 4; ++c) {
        const int col = col0 + c * 16 + l16;
        const float bv = bias[col];
        #pragma unroll
        for (int r = 0; r < 8; ++r) {
            const int m = r + half * 8;
            Y[(size_t)(row0 + m) * n_cols + col] = scale * (accs[c][r] + bv);
        }
    }
}

// One wave per query. Phase 1: lane<->neighbor (scores + softmax via wave32
// shuffles). Phase 2: lane<->dim (coalesced V gather, weights from LDS).
__global__ __launch_bounds__(256) void local_attn(
    const float* __restrict__ qP, const float* __restrict__ kP,
    const float* __restrict__ vP, const long long* __restrict__ idxP,
    float* __restrict__ attn_out, float* __restrict__ wavg, int n_q)
{
    __shared__ float q_lds[8][EMBED];
    __shared__ float w_lds[8][HEADS * KNBR];
    __shared__ int   idx_lds[8][KNBR];

    const int lane = threadIdx.x & 31;
    const int wave = threadIdx.x >> 5;
    int n = blockIdx.x * (blockDim.x >> 5) + wave;
    if (n >= n_q) n = n_q - 1;      // tail waves redo the last query
                                    // (same result; keeps block barriers convergent)

    #pragma unroll
    for (int j = 0; j < EMBED / 32; ++j)
        q_lds[wave][j * 32 + lane] = qP[(size_t)n * EMBED + j * 32 + lane];

    const long long rawidx = idxP[(size_t)n * KNBR + lane];
    const bool valid = rawidx >= 0;
    const int  safe  = valid ? (int)rawidx : 0;
    idx_lds[wave][lane] = safe;
    __syncthreads();

    float wsum = 0.f;
    const float* krow = kP + (size_t)safe * EMBED;
    #pragma unroll
    for (int h = 0; h < HEADS; ++h) {
        float s = 0.f;
        #pragma unroll
        for (int d = 0; d < HDIM; d += 4) {
            float4 kv = *(const float4*)(krow + h * HDIM + d);
            s = fmaf(q_lds[wave][h * HDIM + d + 0], kv.x, s);
            s = fmaf(q_lds[wave][h * HDIM + d + 1], kv.y, s);
            s = fmaf(q_lds[wave][h * HDIM + d + 2], kv.z, s);
            s = fmaf(q_lds[wave][h * HDIM + d + 3], kv.w, s);
        }
        s = valid ? s : -INFINITY;

        float m = s;
        #pragma unroll
        for (int off = 16; off > 0; off >>= 1)
            m = fmaxf(m, __shfl_xor(m, off, 32));
        const float e = valid ? expf(s - m) : 0.f;
        float se = e;
        #pragma unroll
        for (int off = 16; off > 0; off >>= 1)
            se += __shfl_xor(se, off, 32);
        const float a = e / se;           // all-invalid row -> NaN, like reference
        w_lds[wave][h * KNBR + lane] = a;
        wsum += a;
    }
    __syncthreads();

    #pragma unroll
    for (int h = 0; h < HEADS; ++h) {
        float acc = 0.f;
        #pragma unroll 8
        for (int k = 0; k < KNBR; ++k) {
            const float a = w_lds[wave][h * KNBR + k];
            acc = fmaf(a, vP[(size_t)idx_lds[wave][k] * EMBED + h * HDIM + lane], acc);
        }
        attn_out[(size_t)n * EMBED + h * HDIM + lane] = acc;
    }
    wavg[(size_t)n * KNBR + lane] = wsum * (1.0f / HEADS);
}

extern "C" void kernel_launch(void* const* d_in, const int* in_sizes, int n_in,
                              void* d_out, int out_size, void* d_ws, size_t ws_size,
                              hipStream_t stream) {
    const float*     query = (const float*)d_in[0];
    const float*     key   = (const float*)d_in[1];
    const float*     value = (const float*)d_in[2];
    const float*     ipw   = (const float*)d_in[3];   // (3*E, E)
    const float*     ipb   = (const float*)d_in[4];   // (3*E,)
    const float*     opw   = (const float*)d_in[5];   // (E, E)
    const float*     opb   = (const float*)d_in[6];   // (E,)
    const long long* idx   = (const long long*)d_in[7];

    const int n_q = in_sizes[0] / EMBED;
    const int n_k = in_sizes[1] / EMBED;

    float* qws = (float*)d_ws;
    float* kws = qws + (size_t)n_q * EMBED;
    float* vws = kws + (size_t)n_k * EMBED;
    float* aws = vws + (size_t)n_k * EMBED;

    float* out  = (float*)d_out;
    float* wavg = out + (size_t)n_q * EMBED;

    const float scaling = 0.17677669529663687f;       // (EMBED/HEADS)^-0.5

    const dim3 blk(256);
    auto gemm_grid = [](int rows) {
        const int colgroups = EMBED / 64;
        const int mgroups   = ((rows + 15) / 16 + 7) / 8;   // 8 M-tiles per block
        return dim3(colgroups * mgroups);
    };

    gemm_xwt_wmma<<<gemm_grid(n_q), blk, 0, stream>>>(query, ipw,                   ipb,             qws, n_q, EMBED, EMBED, scaling);
    gemm_xwt_wmma<<<gemm_grid(n_k), blk, 0, stream>>>(key,   ipw +     EMBED*EMBED, ipb +     EMBED, kws, n_k, EMBED, EMBED, 1.0f);
    gemm_xwt_wmma<<<gemm_grid(n_k), blk, 0, stream>>>(value, ipw + 2*EMBED*EMBED,   ipb + 2*EMBED,   vws, n_k, EMBED, EMBED, 1.0f);

    local_attn<<<dim3((n_q + 7) / 8), blk, 0, stream>>>(qws, kws, vws, idx, aws, wavg, n_q);

    gemm_xwt_wmma<<<gemm_grid(n_q), blk, 0, stream>>>(aws, opw, opb, out, n_q, EMBED, EMBED, 1.0f);
}